// AdvancedIQCNN_3865470567183
// MI455X (gfx1250) — compile-verified
//
#include <hip/hip_runtime.h>
#include <hip/hip_bf16.h>
#include <math.h>

#define NQ    13
#define DIM   8192           // 2^13
#define PMASK (DIM - 2)      // 0x1FFE: bits 1..12 for CX-chain perm
#define BATCH 4096
#define F1    26             // layer-1 width
#define F2    13             // layer-2 width (= NQ angles)

typedef __attribute__((ext_vector_type(2))) float v2f;
typedef __attribute__((ext_vector_type(8))) float v8f;

// ---------------------------------------------------------------------------
// CX-chain permutation closed form: composing p_c[j] = j ^ (((j>>c)&1)<<(c+1))
// for c = n-2..0 yields bit_t' = bit_t ^ bit_{t-1}  =>  j ^ ((j<<1) & 0x1FFE).
// new_state[j] = old_state[cx_perm(j)].
// ---------------------------------------------------------------------------
__device__ __forceinline__ int cx_perm(int j) { return j ^ ((j << 1) & PMASK); }

// ===========================================================================
// K1: z1 = x @ W1^T + b1   (4096x13 @ 13x26 -> 4096x26) via WMMA f32 16x16x4.
// One wave per 16x16 output tile; 256 M-tiles x 2 N-tiles = 512 waves.
// A(16x4): lane m(0-15) v0=K0,v1=K1; lane m+16 v0=K2,v1=K3.
// B(4x16): lane n(0-15) v0=K0,v1=K1; lane n+16 v0=K2,v1=K3.
// C/D(16x16): vgpr r, lanes0-15 -> M=r, lanes16-31 -> M=8+r; N = lane%16.
// Guards use clamped addresses + value select (no exec-mask branches).
// ===========================================================================
__global__ __launch_bounds__(256) void iq_k1_gemm(
    const float* __restrict__ x, const float* __restrict__ W1,
    const float* __restrict__ b1, float* __restrict__ z1) {
  const int wave = (blockIdx.x * blockDim.x + threadIdx.x) >> 5;
  const int lane = threadIdx.x & 31;
  const int mt = wave >> 1;        // 0..255
  const int nt = wave & 1;         // 0..1
  const int half = lane >> 4;      // 0/1
  const int l16 = lane & 15;
  const int m = mt * 16 + l16;                 // A row (< 4096 always)
  const int n = nt * 16 + l16;                 // B col (pad beyond 26)
  const int nc = (n < F1) ? n : (F1 - 1);      // clamped W1 row

  v8f acc = {};
#pragma unroll
  for (int ks = 0; ks < 4; ++ks) {             // K = 13 padded to 16
    const int k0 = ks * 4 + 2 * half;
    const int kc0 = (k0 < F2) ? k0 : (F2 - 1);
    const int kc1 = (k0 + 1 < F2) ? (k0 + 1) : (F2 - 1);
    v2f a, b;
    const float ax = x[m * F2 + kc0], ay = x[m * F2 + kc1];
    const float bx = W1[nc * F2 + kc0], by = W1[nc * F2 + kc1];
    a.x = (k0     < F2) ? ax : 0.f;
    a.y = (k0 + 1 < F2) ? ay : 0.f;
    b.x = (n < F1 && k0     < F2) ? bx : 0.f;
    b.y = (n < F1 && k0 + 1 < F2) ? by : 0.f;
    acc = __builtin_amdgcn_wmma_f32_16x16x4_f32(
        false, a, false, b, (short)0, acc, false, false);
  }
  const float bias = b1[nc];
#pragma unroll
  for (int r = 0; r < 8; ++r) {
    const int mm = mt * 16 + r + 8 * half;
    const int nn = nt * 16 + l16;
    if (nn < F1) z1[mm * F1 + nn] = acc[r] + bias;
  }
}

// ===========================================================================
// Batch-norm stats: single block, fixed-tree, two-pass (deterministic).
// stats[2f]=mean, stats[2f+1]=rsqrt(var+1e-5)
// ===========================================================================
__global__ __launch_bounds__(256) void iq_stats(
    const float* __restrict__ z, int rows, int cols, float* __restrict__ stats) {
  __shared__ float red[256];
  const int t = threadIdx.x;
  for (int f = 0; f < cols; ++f) {
    float s = 0.f;
    for (int r = t; r < rows; r += 256) s += z[r * cols + f];
    red[t] = s; __syncthreads();
    for (int o = 128; o > 0; o >>= 1) {
      if (t < o) red[t] += red[t + o];
      __syncthreads();
    }
    const float mu = red[0] / (float)rows;
    __syncthreads();
    float q = 0.f;
    for (int r = t; r < rows; r += 256) {
      const float d = z[r * cols + f] - mu;
      q += d * d;
    }
    red[t] = q; __syncthreads();
    for (int o = 128; o > 0; o >>= 1) {
      if (t < o) red[t] += red[t + o];
      __syncthreads();
    }
    if (t == 0) {
      stats[2 * f]     = mu;
      stats[2 * f + 1] = rsqrtf(red[0] / (float)rows + 1e-5f);
    }
    __syncthreads();
  }
}

// ===========================================================================
// K3: h1 = relu(bn(z1)); z2 = h1 @ W2^T + b2   (per-row, tiny)
// ===========================================================================
__global__ __launch_bounds__(256) void iq_k3(
    const float* __restrict__ z1, const float* __restrict__ stats1,
    const float* __restrict__ g1, const float* __restrict__ beta1,
    const float* __restrict__ W2, const float* __restrict__ b2,
    float* __restrict__ z2) {
  const int r = blockIdx.x * blockDim.x + threadIdx.x;
  if (r >= BATCH) return;
  float h1[F1];
#pragma unroll
  for (int f = 0; f < F1; ++f) {
    float v = z1[r * F1 + f];
    v = (v - stats1[2 * f]) * stats1[2 * f + 1] * g1[f] + beta1[f];
    h1[f] = fmaxf(v, 0.f);
  }
#pragma unroll
  for (int n = 0; n < F2; ++n) {
    float s = b2[n];
#pragma unroll
    for (int k = 0; k < F1; ++k) s += W2[n * F1 + k] * h1[k];
    z2[r * F2 + n] = s;
  }
}

// ---------------------------------------------------------------------------
// Quantum helpers.  State lives in LDS; 256 threads (8 waves) per workgroup.
//
// RY layer on a batch-independent theta = Kron of 2x2 rotations.  Qubit
// groups {0-3},{4-7},{8-11} each become a fixed 16x16 matrix U applied along
// one axis of the 8192-state: per group, 512 independent 16-vectors
// -> D(16x16) = U(16x16) * S(16x512) tiles via V_WMMA_F32_16X16X4_F32.
// Qubit 12 is a single stride-4096 butterfly.
//
// Element address for group g, column c (9 bits), k (0..15):
//   addr = ((c >> 4g) << (4g+4)) | (k << 4g) | (c & ((1<<4g)-1))
// Each output column depends only on its own column's inputs and a wave reads
// all 16 k before writing back -> in-place within a group pass, no barrier.
// ---------------------------------------------------------------------------
__device__ __forceinline__ void iq_group_wmma(const float* src, float* dst,
                                              const float* U, int g,
                                              bool fuse_perm) {
  const int wv = threadIdx.x >> 5;
  const int lane = threadIdx.x & 31;
  const int half = lane >> 4;
  const int l16 = lane & 15;

  // A fragments: U[m][k], m = l16; reused across all tiles of this pass.
  v2f afr[4];
#pragma unroll
  for (int ks = 0; ks < 4; ++ks) {
    const int k0 = ks * 4 + 2 * half;
    afr[ks].x = U[l16 * 16 + k0];
    afr[ks].y = U[l16 * 16 + k0 + 1];
  }

  const int shift = 4 * g;
  const int lowmask = (1 << shift) - 1;
  for (int tile = wv; tile < 32; tile += 8) {      // 512 cols / 16 per tile
    const int col = tile * 16 + l16;
    const int base = ((col >> shift) << (shift + 4)) | (col & lowmask);
    v8f acc = {};
#pragma unroll
    for (int ks = 0; ks < 4; ++ks) {
      const int k0 = ks * 4 + 2 * half;
      int a0 = base | (k0 << shift);
      int a1 = base | ((k0 + 1) << shift);
      if (fuse_perm) { a0 = cx_perm(a0); a1 = cx_perm(a1); }
      v2f bfr;
      bfr.x = src[a0];
      bfr.y = src[a1];
      acc = __builtin_amdgcn_wmma_f32_16x16x4_f32(
          false, afr[ks], false, bfr, (short)0, acc, false, false);
    }
#pragma unroll
    for (int r = 0; r < 8; ++r) {
      const int mrow = r + 8 * half;
      dst[base | (mrow << shift)] = acc[r];
    }
  }
}

__device__ __forceinline__ void iq_q12(float* st, float c, float s) {
  for (int p = threadIdx.x; p < DIM / 2; p += 256) {
    const int i1 = p | (DIM / 2);
    const float a0 = st[p], a1 = st[i1];
    st[p]  = c * a0 - s * a1;
    st[i1] = s * a0 + c * a1;
  }
}

// Full RY(theta) layer.  If fuse_perm, the CX-chain permutation is folded
// into the first group pass (reads src permuted, writes dst).
// Caller must ensure a barrier precedes this call.
__device__ __forceinline__ void iq_ry_wmma(const float* src, float* dst,
                                           float (*Uset)[256], float c12,
                                           float s12, bool fuse_perm) {
  iq_group_wmma(src, dst, Uset[0], 0, fuse_perm);
  __syncthreads();
  iq_group_wmma(dst, dst, Uset[1], 1, false);
  __syncthreads();
  iq_group_wmma(dst, dst, Uset[2], 2, false);
  __syncthreads();
  iq_q12(dst, c12, s12);
  __syncthreads();
}

// Product-state init fused with CX-chain gather: after RY(x) on |0>,
// amp(j) = prod_q (bit_q(j) ? s_q : c_q); then permuted.
__device__ __forceinline__ void iq_init_perm(float* st, const float* c0,
                                             const float* s0) {
  for (int j = threadIdx.x; j < DIM; j += 256) {
    const int v = cx_perm(j);
    float amp = 1.f;
#pragma unroll
    for (int q = 0; q < NQ; ++q) amp *= ((v >> q) & 1) ? s0[q] : c0[q];
    st[j] = amp;
  }
}

// p1 = sum over odd j of st[j]^2 (fixed LDS tree, deterministic).
__device__ __forceinline__ float iq_measure(const float* st, float* red) {
  float acc = 0.f;
#pragma unroll
  for (int i = 0; i < DIM / 512; ++i) {
    const int j = ((i * 256 + threadIdx.x) << 1) | 1;
    const float v = st[j];
    acc += v * v;
  }
  red[threadIdx.x] = acc; __syncthreads();
  for (int o = 128; o > 0; o >>= 1) {
    if (threadIdx.x < o) red[threadIdx.x] += red[threadIdx.x + o];
    __syncthreads();
  }
  const float r = red[0];
  __syncthreads();
  return r;
}

// ===========================================================================
// K5: quantum circuits (WMMA RY layers) + head MLP, one WG per batch row.
// LDS: 2x32KB state buffers + 9KB U tables + small  (~75KB -> 4 WGs/WGP)
// ===========================================================================
__global__ __launch_bounds__(256) void iq_quantum(
    const float* __restrict__ z2, const float* __restrict__ stats2,
    const float* __restrict__ g2, const float* __restrict__ beta2,
    const float* __restrict__ th1a, const float* __restrict__ th1b,
    const float* __restrict__ th2a,
    const float* __restrict__ W3, const float* __restrict__ b3,
    const float* __restrict__ W4, const float* __restrict__ b4,
    const float* __restrict__ W5, const float* __restrict__ b5,
    float* __restrict__ out) {
  __shared__ float stA[DIM];
  __shared__ float stB[DIM];
  __shared__ float U[3][3][256];    // [theta set][qubit group][row*16+col]
  __shared__ float cb[4][NQ];
  __shared__ float sb[4][NQ];
  __shared__ float red[256];
  __shared__ float h3[32];
  __shared__ float h4[16];
  const int t = threadIdx.x;
  const int b = blockIdx.x;

  // Trig tables: set 0 = data angles (bn+relu of z2), sets 1..3 = thetas.
  if (t < NQ) {
    const float mu = stats2[2 * t], ri = stats2[2 * t + 1];
    float hv = (z2[b * F2 + t] - mu) * ri * g2[t] + beta2[t];
    hv = fmaxf(hv, 0.f) * 0.5f;
    cb[0][t] = cosf(hv); sb[0][t] = sinf(hv);
  } else if (t >= 32 && t < 32 + NQ) {
    const int q = t - 32; const float hv = th1a[q] * 0.5f;
    cb[1][q] = cosf(hv); sb[1][q] = sinf(hv);
  } else if (t >= 64 && t < 64 + NQ) {
    const int q = t - 64; const float hv = th1b[q] * 0.5f;
    cb[2][q] = cosf(hv); sb[2][q] = sinf(hv);
  } else if (t >= 96 && t < 96 + NQ) {
    const int q = t - 96; const float hv = th2a[q] * 0.5f;
    cb[3][q] = cosf(hv); sb[3][q] = sinf(hv);
  }
  __syncthreads();

  // Build 16x16 group matrices: U[row][col] = prod_i R_{4g+i}[bit_i(row)][bit_i(col)]
  // with R = [[c,-s],[s,c]].  One entry per thread per matrix.
  {
    const int row = t >> 4, colm = t & 15;
#pragma unroll
    for (int set = 0; set < 3; ++set)
#pragma unroll
      for (int g = 0; g < 3; ++g) {
        float u = 1.f;
#pragma unroll
        for (int i = 0; i < 4; ++i) {
          const int q = 4 * g + i;
          const float c = cb[set + 1][q], s = sb[set + 1][q];
          const int ro = (row >> i) & 1, ci = (colm >> i) & 1;
          u *= ro ? (ci ? c : s) : (ci ? -s : c);
        }
        U[set][g][t] = u;
      }
  }

  // Circuit A: init(prod+perm) -> RY(th1a) -> (perm fused) RY(th1b) -> measure
  iq_init_perm(stA, cb[0], sb[0]);
  __syncthreads();
  iq_ry_wmma(stA, stA, U[0], cb[1][NQ - 1], sb[1][NQ - 1], false);
  iq_ry_wmma(stA, stB, U[1], cb[2][NQ - 1], sb[2][NQ - 1], true);
  const float pA = iq_measure(stB, red);

  // Circuit B: init(prod+perm) -> RY(th2a) -> measure
  iq_init_perm(stA, cb[0], sb[0]);
  __syncthreads();
  iq_ry_wmma(stA, stA, U[2], cb[3][NQ - 1], sb[3][NQ - 1], false);
  const float pB = iq_measure(stA, red);

  const float xq = 0.5f * (pA + pB);  // uniform across all threads

  // Head MLP: 1 -> 32 -> 16 -> 2
  if (t < 32) h3[t] = fmaxf(xq * W3[t] + b3[t], 0.f);
  __syncthreads();
  if (t < 16) {
    float s = b4[t];
#pragma unroll
    for (int j = 0; j < 32; ++j) s += W4[t * 32 + j] * h3[j];
    h4[t] = fmaxf(s, 0.f);
  }
  __syncthreads();
  if (t < 2) {
    float s = b5[t];
#pragma unroll
    for (int j = 0; j < 16; ++j) s += W5[t * 16 + j] * h4[j];
    out[b * 2 + t] = s;
  }
}

// ===========================================================================
extern "C" void kernel_launch(void* const* d_in, const int* in_sizes, int n_in,
                              void* d_out, int out_size, void* d_ws,
                              size_t ws_size, hipStream_t stream) {
  const float* x     = (const float*)d_in[0];
  const float* W1    = (const float*)d_in[1];
  const float* b1    = (const float*)d_in[2];
  const float* g1    = (const float*)d_in[3];
  const float* beta1 = (const float*)d_in[4];
  const float* W2    = (const float*)d_in[5];
  const float* b2    = (const float*)d_in[6];
  const float* g2    = (const float*)d_in[7];
  const float* beta2 = (const float*)d_in[8];
  const float* th1a  = (const float*)d_in[9];
  const float* th1b  = (const float*)d_in[10];
  const float* th2a  = (const float*)d_in[11];
  const float* W3    = (const float*)d_in[12];
  const float* b3    = (const float*)d_in[13];
  const float* W4    = (const float*)d_in[14];
  const float* b4    = (const float*)d_in[15];
  const float* W5    = (const float*)d_in[16];
  const float* b5    = (const float*)d_in[17];
  float* out = (float*)d_out;

  // Workspace layout (~660 KB)
  float* z1     = (float*)d_ws;          // 4096*26
  float* z2     = z1 + BATCH * F1;       // 4096*13
  float* stats1 = z2 + BATCH * F2;       // 2*26
  float* stats2 = stats1 + 2 * F1;       // 2*13

  iq_k1_gemm<<<64, 256, 0, stream>>>(x, W1, b1, z1);
  iq_stats<<<1, 256, 0, stream>>>(z1, BATCH, F1, stats1);
  iq_k3<<<BATCH / 256, 256, 0, stream>>>(z1, stats1, g1, beta1, W2, b2, z2);
  iq_stats<<<1, 256, 0, stream>>>(z2, BATCH, F2, stats2);
  iq_quantum<<<BATCH, 256, 0, stream>>>(z2, stats2, g2, beta2, th1a, th1b,
                                        th2a, W3, b3, W4, b4, W5, b5, out);
}